// SAGEConv_14851996909836
// MI455X (gfx1250) — compile-verified
//
#include <hip/hip_runtime.h>
#include <math.h>

typedef float v2f __attribute__((ext_vector_type(2)));
typedef float v8f __attribute__((ext_vector_type(8)));

#define LL 2
#define NN 50000
#define CC 128
#define LDW 132                     // LDS row stride (pad: 4*o bank pattern, 8B aligned)
#define LDS_BYTES (2 * CC * LDW * 4)
#define LN_EPS 1e-5f

// ---------------- Kernel 1: zero agg workspace ----------------
__global__ void zero_ws_kernel(float4* __restrict__ p, long n4) {
    long i = (long)blockIdx.x * blockDim.x + threadIdx.x;
    if (i < n4) p[i] = make_float4(0.f, 0.f, 0.f, 0.f);
}

// ---------------- Kernel 2: edge scatter  agg[src] += val * x[dst] ----------------
__global__ void scatter_edges_kernel(const float* __restrict__ x,
                                     const int*   __restrict__ esrc,
                                     const int*   __restrict__ edst,
                                     const float* __restrict__ eval,
                                     float*       __restrict__ agg,
                                     int nedges) {
    const int e    = (int)(((long)blockIdx.x * blockDim.x + threadIdx.x) >> 5);
    const int lane = threadIdx.x & 31;
    if (e >= nedges) return;
    const int   s = esrc[e];
    const int   d = edst[e];
    const float v = eval[e];
    const int   c = lane * 4;                       // 32 lanes x 4 = 128 channels
#pragma unroll
    for (int l = 0; l < LL; ++l) {
        const float4 xv = *(const float4*)(x + ((size_t)l * NN + d) * CC + c);
        float* ap = agg + ((size_t)l * NN + s) * CC + c;
        atomicAdd(ap + 0, v * xv.x);
        atomicAdd(ap + 1, v * xv.y);
        atomicAdd(ap + 2, v * xv.z);
        atomicAdd(ap + 3, v * xv.w);
    }
}

// ---------------- Kernel 3: fused dual-GEMM + bias + SiLU + LayerNorm ----------------
// grid: (ceil(N/128), L), block: 256 (8 waves). Wave w owns rows [base+16w, base+16w+16).
__global__ void fused_gemm_silu_ln_kernel(const float* __restrict__ x,
                                          const float* __restrict__ agg,
                                          const float* __restrict__ W,
                                          const float* __restrict__ bvec,
                                          const float* __restrict__ Wagg,
                                          const float* __restrict__ gamma,
                                          const float* __restrict__ beta,
                                          float*       __restrict__ out) {
    extern __shared__ float lds[];                  // [2][CC*LDW]
    float* ldsW  = lds;
    float* ldsWa = lds + CC * LDW;

    const int tid = threadIdx.x;
    const int l   = blockIdx.y;

    // Stage W and W_agg into LDS (row-major, padded stride LDW)
    for (int i = tid; i < CC * CC; i += blockDim.x) {
        const int o = i >> 7, c = i & (CC - 1);
        ldsW [o * LDW + c] = W[i];
        ldsWa[o * LDW + c] = Wagg[i];
    }
    __syncthreads();

    const int wave  = tid >> 5;
    const int lane  = tid & 31;
    const int half  = lane >> 4;                    // 0: lanes 0-15, 1: lanes 16-31
    const int lr    = lane & 15;
    const int kbase = half * 2;                     // A/B frag K split per ISA layout

    const int rowBase = blockIdx.x * 128 + wave * 16;
    int arow = rowBase + lr;                        // A-matrix row for this lane
    if (arow >= NN) arow = NN - 1;                  // clamp: EXEC must stay all-1s for WMMA
    const float* xrow = x   + ((size_t)l * NN + arow) * CC;
    const float* grow = agg + ((size_t)l * NN + arow) * CC;

    v8f acc[8] = {};                                // 8 col-tiles x 16x16 f32 C/D

    for (int k0 = 0; k0 < CC; k0 += 4) {
        const v2f ax = *(const v2f*)(xrow + k0 + kbase);
        const v2f ag = *(const v2f*)(grow + k0 + kbase);
#pragma unroll
        for (int t = 0; t < 8; ++t) {
            const int o = t * 16 + lr;              // output column for B frag
            const v2f bw = *(const v2f*)(&ldsW [o * LDW + k0 + kbase]);
            const v2f ba = *(const v2f*)(&ldsWa[o * LDW + k0 + kbase]);
            acc[t] = __builtin_amdgcn_wmma_f32_16x16x4_f32(false, ax, false, bw,
                                                           (short)0, acc[t], false, false);
            acc[t] = __builtin_amdgcn_wmma_f32_16x16x4_f32(false, ag, false, ba,
                                                           (short)0, acc[t], false, false);
        }
    }

    // Per-column constants for this lane
    float bc[8], gc[8], btc[8];
#pragma unroll
    for (int t = 0; t < 8; ++t) {
        const int col = t * 16 + lr;
        bc[t]  = bvec[col];
        gc[t]  = gamma[col];
        btc[t] = beta[col];
    }

    // Epilogue: bias + SiLU, then LayerNorm per row.
    // C/D layout: VGPR j holds row (rowBase+j) in lanes 0-15 and row (rowBase+j+8)
    // in lanes 16-31, columns = 16t + lr. A 16-lane butterfly (masks 1..8) reduces
    // each half independently = one full row of 128 values.
#pragma unroll
    for (int j = 0; j < 8; ++j) {
        float s = 0.f, s2 = 0.f;
#pragma unroll
        for (int t = 0; t < 8; ++t) {
            const float p = acc[t][j] + bc[t];
            const float v = p / (1.f + __expf(-p));   // SiLU
            acc[t][j] = v;
            s  += v;
            s2 += v * v;
        }
#pragma unroll
        for (int m = 1; m <= 8; m <<= 1) {
            s  += __shfl_xor(s,  m, 32);
            s2 += __shfl_xor(s2, m, 32);
        }
        const float mean = s  * (1.f / 128.f);
        const float var  = s2 * (1.f / 128.f) - mean * mean;
        const float inv  = rsqrtf(var + LN_EPS);

        const int row = rowBase + j + half * 8;
        if (row < NN) {
            float* orow = out + ((size_t)l * NN + row) * CC;
#pragma unroll
            for (int t = 0; t < 8; ++t)
                orow[t * 16 + lr] = (acc[t][j] - mean) * inv * gc[t] + btc[t];
        }
    }
}

extern "C" void kernel_launch(void* const* d_in, const int* in_sizes, int n_in,
                              void* d_out, int out_size, void* d_ws, size_t ws_size,
                              hipStream_t stream) {
    const float* x     = (const float*)d_in[0];
    const int*   esrc  = (const int*)  d_in[1];
    const int*   edst  = (const int*)  d_in[2];
    const float* eval  = (const float*)d_in[3];
    const float* W     = (const float*)d_in[4];
    const float* b     = (const float*)d_in[5];
    const float* Wagg  = (const float*)d_in[6];
    const float* gamma = (const float*)d_in[7];
    const float* beta  = (const float*)d_in[8];
    float*       out   = (float*)d_out;
    float*       agg   = (float*)d_ws;              // [L, N, C] fp32 = 51.2 MB
    const int    nedges = in_sizes[1];

    // 1) zero agg
    const long n4 = (long)LL * NN * CC / 4;
    zero_ws_kernel<<<(int)((n4 + 255) / 256), 256, 0, stream>>>((float4*)agg, n4);

    // 2) edge scatter (one wave32 per edge)
    const int edges_per_block = 256 / 32;
    scatter_edges_kernel<<<(nedges + edges_per_block - 1) / edges_per_block, 256, 0, stream>>>(
        x, esrc, edst, eval, agg, nedges);

    // 3) fused dual-GEMM + SiLU + LayerNorm
    dim3 grid((NN + 127) / 128, LL);
    fused_gemm_silu_ln_kernel<<<grid, 256, LDS_BYTES, stream>>>(
        x, agg, W, b, Wagg, gamma, beta, out);
}